// DeepseekV4SWACache_53455162966095
// MI455X (gfx1250) — compile-verified
//
#include <hip/hip_runtime.h>
#include <hip/hip_bf16.h>
#include <stdint.h>

// Problem constants (fixed by the reference's setup_inputs).
#define NUM_DECODES   256
#define NUM_PREFILLS  8
#define WINDOW        512
#define BLKSZ         64
#define MAX_BLOCKS    32
#define HEAD          256
#define NROWS         (NUM_DECODES * WINDOW)        // 131072 gather rows

// d_out layout in 4-byte elements (tuple concatenated flat, in return order):
//   swa_indices [256*512] i32 | swa_lens [256] i32 | prefill_gather_lens [8] i32 | gathered f32
#define OFF_IDX   0
#define OFF_LENS  (NUM_DECODES * WINDOW)            // 131072
#define OFF_PF    (OFF_LENS + NUM_DECODES)          // 131328
#define OFF_GATH  (OFF_PF + NUM_PREFILLS)           // 131336 (x4B = 16B aligned)

#define WAVES_PER_BLOCK 4
#define GROUP           4                            // rows per pipeline stage
#define NGROUPS         4
#define ROWS_PER_WAVE   (GROUP * NGROUPS)            // 16 (always within one token)
#define GRID_B          (NROWS / (WAVES_PER_BLOCK * ROWS_PER_WAVE))  // 2048

typedef float v4f __attribute__((ext_vector_type(4)));

// ---------------------------------------------------------------------------
// Kernel A: per-decode-token window metadata + prefill gather lengths.
// ---------------------------------------------------------------------------
__global__ void swa_setup_kernel(const int* __restrict__ seq_lens,
                                 const int* __restrict__ qsl,
                                 const int* __restrict__ t2r,
                                 const int* __restrict__ slot_mapping,
                                 int* __restrict__ swa_lens_out,
                                 int* __restrict__ pf_out,
                                 int* __restrict__ ws_start,
                                 int* __restrict__ ws_len,
                                 int* __restrict__ ws_req) {
  int t = blockIdx.x * blockDim.x + threadIdx.x;
  if (t < NUM_DECODES) {
    int req  = t2r[t];
    int qs   = qsl[req];
    int qlen = qsl[req + 1] - qs;
    int seq  = seq_lens[req];
    int pos  = (seq - qlen) + t - qs;          // decode: == seq-1
    int start = pos - (WINDOW - 1);
    if (start < 0) start = 0;
    int len = pos + 1 - start;                 // == min(pos+1, WINDOW)
    if (slot_mapping[t] < 0) len = 0;
    swa_lens_out[t] = len;
    ws_start[t] = start;
    ws_len[t]   = len;
    ws_req[t]   = req;
  } else if (t < NUM_DECODES + NUM_PREFILLS) {
    int p    = t - NUM_DECODES;
    int i    = NUM_DECODES + p;
    int seq  = seq_lens[i];
    int qlen = qsl[i + 1] - qsl[i];
    int prefix = seq - qlen;
    int m = prefix < (WINDOW - 1) ? prefix : (WINDOW - 1);
    pf_out[p] = qlen + m;
  }
}

// ---------------------------------------------------------------------------
// Kernel B: 128MB gather via CDNA5 async global<->LDS DMA, grouped pipeline.
// Per wave: 4 groups of 4 rows; double-buffered 4KB LDS sets. Each group phase
// keeps 8 async-loads + 8 async-stores (8KB) in flight between asynccnt waits.
// ---------------------------------------------------------------------------
__device__ __forceinline__ unsigned lds_byte_off(const void* p) {
  // Flat shared pointer = {SHARED_BASE, lds_offset}; low 32 bits = LDS byte addr.
  return (unsigned)(uintptr_t)p;
}

__device__ __forceinline__ void async_row_load(unsigned l, const float* p) {
  asm volatile("global_load_async_to_lds_b128 %0, %1, off th:TH_LOAD_NT"
               :: "v"(l), "v"(p) : "memory");
  asm volatile("global_load_async_to_lds_b128 %0, %1, off offset:512 th:TH_LOAD_NT"
               :: "v"(l), "v"(p) : "memory");
}

__device__ __forceinline__ void async_row_store(float* d, unsigned l) {
  asm volatile("global_store_async_from_lds_b128 %0, %1, off th:TH_STORE_NT"
               :: "v"(d), "v"(l) : "memory");
  asm volatile("global_store_async_from_lds_b128 %0, %1, off offset:512 th:TH_STORE_NT"
               :: "v"(d), "v"(l) : "memory");
}

__global__ __launch_bounds__(WAVES_PER_BLOCK * 32)
void swa_gather_kernel(const float* __restrict__ kv,
                       const int*   __restrict__ block_table,
                       const int*   __restrict__ ws_start,
                       const int*   __restrict__ ws_len,
                       const int*   __restrict__ ws_req,
                       int*   __restrict__ idx_out,
                       float* __restrict__ gathered) {
  __shared__ float lds[WAVES_PER_BLOCK * 2 * GROUP * HEAD];  // 32KB/block

  const int wave = threadIdx.x >> 5;
  const int lane = threadIdx.x & 31;
  const unsigned waveBase = lds_byte_off(&lds[wave * 2 * GROUP * HEAD]);
  const unsigned laneB    = (unsigned)lane * 16u;
  const int rowBase = (blockIdx.x * WAVES_PER_BLOCK + wave) * ROWS_PER_WAVE;

  // All 16 rows of this wave share one token (16 divides 512).
  const int t     = rowBase >> 9;
  const int start = ws_start[t];
  const int len   = ws_len[t];
  const int req   = ws_req[t];
  const int off0  = rowBase & (WINDOW - 1);

  // Per-row metadata: src row pointer, slot index, in-window flag.
  auto row_meta = [&](int gj, const float*& s, int& iv) -> bool {
    const int off = off0 + gj;
    const bool w = off < len;               // wave-uniform
    iv = -1; s = kv;
    if (w) {
      const int pos_off = start + off;
      int bi = pos_off >> 6;
      if (bi > MAX_BLOCKS - 1) bi = MAX_BLOCKS - 1;  // JAX clamps OOB block_table gather
      const int blk  = block_table[req * MAX_BLOCKS + bi];
      const int slot = blk * BLKSZ + (pos_off & (BLKSZ - 1));
      iv = slot;
      s  = kv + (size_t)slot * HEAD;
    }
    return w;
  };

  const float* curS[GROUP]; bool curW[GROUP]; int curI[GROUP];

  // Prologue: group 0 loads into set 0.
  {
    const unsigned setB = waveBase;
#pragma unroll
    for (int j = 0; j < GROUP; ++j) {
      curW[j] = row_meta(j, curS[j], curI[j]);
      if (curW[j]) async_row_load(setB + (unsigned)j * (HEAD * 4) + laneB, curS[j] + lane * 4);
    }
  }

#pragma unroll
  for (int g = 0; g < NGROUPS; ++g) {
    // Drain: group g's loads landed in LDS; group g-1's stores done (buffer-set
    // (g+1)&1 — last read by stores of g-1 — is now reusable).
    asm volatile("s_wait_asynccnt 0x0" ::: "memory");

    const float* nS[GROUP] = {}; bool nW[GROUP] = {}; int nI[GROUP] = {};
    if (g + 1 < NGROUPS) {
      const unsigned setB = waveBase + (unsigned)((g + 1) & 1) * (GROUP * HEAD * 4);
#pragma unroll
      for (int j = 0; j < GROUP; ++j) {
        nW[j] = row_meta((g + 1) * GROUP + j, nS[j], nI[j]);
        if (nW[j]) async_row_load(setB + (unsigned)j * (HEAD * 4) + laneB, nS[j] + lane * 4);
      }
    }

    // Stores for group g (overlap with group g+1's loads until next wait).
    {
      const unsigned setB = waveBase + (unsigned)(g & 1) * (GROUP * HEAD * 4);
#pragma unroll
      for (int j = 0; j < GROUP; ++j) {
        const int row = rowBase + g * GROUP + j;
        float* gout = gathered + (size_t)row * HEAD;
        if (curW[j]) {
          async_row_store(gout + lane * 4, setB + (unsigned)j * (HEAD * 4) + laneB);
        } else {
          const v4f z = {0.f, 0.f, 0.f, 0.f};
          __builtin_nontemporal_store(z, (v4f*)(gout + lane * 4));
          __builtin_nontemporal_store(z, (v4f*)(gout + 128 + lane * 4));
        }
        if (lane == 0) idx_out[row] = curI[j];
      }
    }

#pragma unroll
    for (int j = 0; j < GROUP; ++j) { curS[j] = nS[j]; curW[j] = nW[j]; curI[j] = nI[j]; }
  }
  // s_endpgm performs an implicit wait-idle, draining remaining ASYNCcnt/STOREcnt.
}

// ---------------------------------------------------------------------------
extern "C" void kernel_launch(void* const* d_in, const int* in_sizes, int n_in,
                              void* d_out, int out_size, void* d_ws, size_t ws_size,
                              hipStream_t stream) {
  const float* kv          = (const float*)d_in[0];
  const int*   seq_lens    = (const int*)d_in[1];
  const int*   qsl         = (const int*)d_in[2];
  const int*   t2r         = (const int*)d_in[3];
  const int*   slot_map    = (const int*)d_in[4];
  const int*   block_table = (const int*)d_in[5];

  int*   outI = (int*)d_out;
  float* outF = (float*)d_out;
  int*   ws   = (int*)d_ws;     // [0,256): start  [256,512): len  [512,768): req

  swa_setup_kernel<<<2, 256, 0, stream>>>(
      seq_lens, qsl, t2r, slot_map,
      outI + OFF_LENS, outI + OFF_PF,
      ws, ws + NUM_DECODES, ws + 2 * NUM_DECODES);

  swa_gather_kernel<<<GRID_B, WAVES_PER_BLOCK * 32, 0, stream>>>(
      kv, block_table,
      ws, ws + NUM_DECODES, ws + 2 * NUM_DECODES,
      outI + OFF_IDX, outF + OFF_GATH);
}